// SMPLXModel_18760417149407
// MI455X (gfx1250) — compile-verified
//
#include <hip/hip_runtime.h>
#include <stdint.h>

// ---------------------------------------------------------------------------
// batch_rigid_transform for SMPL-X (55 joints), BATCH=65536, fp32.
// Memory-bound (~173 MB => ~7.4us roofline @ 23.3 TB/s). TDM stages strided
// [B,55,3,3] tiles into LDS (double-buffered, overlapped with compute);
// per-thread VALU fp32 composes the chain fully unrolled in registers;
// outputs staged in LDS and drained by one contiguous tensor_store_from_lds.
// LDS is accessed through explicit addrspace(3) volatile pointers so the
// backend emits ds_load/ds_store (InferAddressSpaces skips volatile ops).
// ---------------------------------------------------------------------------

#define N_JOINTS 55
#define TILE     64          // batch elements per block (== blockDim.x)
#define CH       5           // joints per staging chunk
#define NSTAGE   11          // 55 / CH

static constexpr int kPar[N_JOINTS] = {
    -1, 0, 0, 0, 1, 2, 3, 4, 5, 6, 7, 8, 9, 9, 9, 12, 13, 14, 16, 17,
    18, 19, 15, 15, 15, 20, 25, 26, 20, 28, 29, 20, 31, 32, 20, 34, 35,
    20, 37, 38, 21, 40, 41, 21, 43, 44, 21, 46, 47, 21, 49, 50, 21, 52, 53};

typedef unsigned int u32x4 __attribute__((ext_vector_type(4)));
typedef int          i32x4 __attribute__((ext_vector_type(4)));
typedef int          i32x8 __attribute__((ext_vector_type(8)));

// Explicit LDS (addrspace 3) volatile pointer type: guarantees ds_* encoding.
typedef volatile __attribute__((address_space(3))) float vldsf;

#if __has_builtin(__builtin_amdgcn_tensor_load_to_lds) && \
    __has_builtin(__builtin_amdgcn_tensor_store_from_lds) && \
    __has_builtin(__builtin_amdgcn_s_wait_tensorcnt)
#define USE_TDM 1
#else
#define USE_TDM 0
#endif

#if USE_TDM
// ---- Tensor DMA Descriptor (D#) packing per CDNA5 ISA §8 -------------------
__device__ __forceinline__ u32x4 tdm_g0(uint32_t lds_byte_off, uint64_t ga) {
  u32x4 g;
  g.x = 1u;                                   // count=1 (valid), user mode
  g.y = lds_byte_off;                         // lds_addr
  g.z = (uint32_t)ga;                         // global_addr[31:0]
  g.w = (uint32_t)((ga >> 32) & 0x01FFFFFFull) | (2u << 30); // addr hi | type=2
  return g;
}

__device__ __forceinline__ i32x8 tdm_g1(uint32_t tensor_d0, uint32_t tensor_d1,
                                        uint32_t tile_d0,  uint32_t tile_d1,
                                        uint32_t stride_d0) {
  i32x8 g;
  g[0] = (int)(2u << 16);                                     // data_size=4B
  g[1] = (int)((tensor_d0 & 0xFFFFu) << 16);                  // dim0[15:0]
  g[2] = (int)(((tensor_d0 >> 16) & 0xFFFFu) |
               ((tensor_d1 & 0xFFFFu) << 16));                // dim0 hi | dim1 lo
  g[3] = (int)(((tensor_d1 >> 16) & 0xFFFFu) |
               ((tile_d0 & 0xFFFFu) << 16));                  // dim1 hi | tile0
  g[4] = (int)(tile_d1 & 0xFFFFu);                            // tile1, tile2=0
  g[5] = (int)stride_d0;                                      // dim0_stride lo
  g[6] = 0;
  g[7] = 0;
  return g;
}

__device__ __forceinline__ void tdm_load(u32x4 g0, i32x8 g1) {
  i32x4 z4 = {0, 0, 0, 0};
#if defined(__clang_major__) && (__clang_major__ >= 23)
  i32x8 z8 = {0, 0, 0, 0, 0, 0, 0, 0};
  __builtin_amdgcn_tensor_load_to_lds(g0, g1, z4, z4, z8, 0);
#else
  __builtin_amdgcn_tensor_load_to_lds(g0, g1, z4, z4, 0);
#endif
}

__device__ __forceinline__ void tdm_store(u32x4 g0, i32x8 g1) {
  i32x4 z4 = {0, 0, 0, 0};
#if defined(__clang_major__) && (__clang_major__ >= 23)
  i32x8 z8 = {0, 0, 0, 0, 0, 0, 0, 0};
  __builtin_amdgcn_tensor_store_from_lds(g0, g1, z4, z4, z8, 0);
#else
  __builtin_amdgcn_tensor_store_from_lds(g0, g1, z4, z4, 0);
#endif
}
#endif  // USE_TDM

__global__ __launch_bounds__(TILE) void smplx_rigid_chain_kernel(
    const float* __restrict__ rot,   // [B, 55, 3, 3]
    const float* __restrict__ Jg,    // [55, 3]
    float* __restrict__ out) {       // [B, 55, 3]
  __shared__ float sIn[2][CH * 9 * TILE];      // 2 x 11.25 KB
  __shared__ float sOut[N_JOINTS * 3 * TILE];  // 41.25 KB  (total 63.75 KB)

  const int tid = (int)threadIdx.x;   // element within tile
  const int blk = (int)blockIdx.x;

  // 32-bit LDS byte offsets (low bits of the flat address) -> AS3 pointers.
  const uint32_t ldsIn0 = (uint32_t)(uintptr_t)&sIn[0][0];
  const uint32_t ldsIn1 = (uint32_t)(uintptr_t)&sIn[1][0];
  const uint32_t ldsOut = (uint32_t)(uintptr_t)&sOut[0];
  vldsf* const pIn0 = (vldsf*)ldsIn0;
  vldsf* const pIn1 = (vldsf*)ldsIn1;
  vldsf* const pOut = (vldsf*)ldsOut;

#if USE_TDM
  const uint64_t rotBase = (uint64_t)(uintptr_t)rot + (uint64_t)blk * (TILE * 1980ull);
  const uint64_t outBase = (uint64_t)(uintptr_t)out + (uint64_t)blk * (TILE * 660ull);

  // Prologue: prefetch stages 0 and 1 (double buffer). In-order completion.
  if (threadIdx.x < 32) {
    tdm_load(tdm_g0(ldsIn0, rotBase + 0u * 180u), tdm_g1(45u, TILE, 45u, TILE, 495u));
    tdm_load(tdm_g0(ldsIn1, rotBase + 1u * 180u), tdm_g1(45u, TILE, 45u, TILE, 495u));
  }
#endif

  float Gr[N_JOINTS][9];   // accumulated rotations (register-promoted; ~5 live)
  float Gt[N_JOINTS][3];   // accumulated translations

#pragma unroll
  for (int s = 0; s < NSTAGE; ++s) {
    vldsf* const pBuf = (s & 1) ? pIn1 : pIn0;
#if USE_TDM
    if (threadIdx.x < 32) {
      // Oldest outstanding load is stage s; allow stage s+1 to stay in flight.
      if (s < NSTAGE - 1) __builtin_amdgcn_s_wait_tensorcnt(1);
      else                __builtin_amdgcn_s_wait_tensorcnt(0);
    }
    __syncthreads();  // buf[s&1] ready for all waves
#else
    __syncthreads();
    for (int t = tid; t < CH * 9 * TILE; t += TILE) {
      const int e = t / 45, o = t % 45;
      pBuf[t] = rot[(size_t)(blk * TILE + e) * 495u + (size_t)(45 * s) + (size_t)o];
    }
    __syncthreads();
#endif

#pragma unroll
    for (int j = 0; j < CH; ++j) {
      const int i = s * CH + j;     // constant after unroll
      const int p = kPar[i];        // constant-folded

      // Conflict-free LDS reads: row stride 45 dwords, gcd(45,64)=1.
      vldsf* const R = pBuf + (tid * 45 + j * 9);
      const float r0 = R[0], r1 = R[1], r2 = R[2];
      const float r3 = R[3], r4 = R[4], r5 = R[5];
      const float r6 = R[6], r7 = R[7], r8 = R[8];

      // rel joint offset (uniform across lanes -> scalar loads).
      float rx, ry, rz;
      if (p < 0) {
        rx = Jg[0]; ry = Jg[1]; rz = Jg[2];
      } else {
        rx = Jg[3 * i + 0] - Jg[3 * p + 0];
        ry = Jg[3 * i + 1] - Jg[3 * p + 1];
        rz = Jg[3 * i + 2] - Jg[3 * p + 2];
      }

      if (p < 0) {
        Gr[i][0] = r0; Gr[i][1] = r1; Gr[i][2] = r2;
        Gr[i][3] = r3; Gr[i][4] = r4; Gr[i][5] = r5;
        Gr[i][6] = r6; Gr[i][7] = r7; Gr[i][8] = r8;
        Gt[i][0] = rx; Gt[i][1] = ry; Gt[i][2] = rz;
      } else {
#pragma unroll
        for (int rr = 0; rr < 3; ++rr) {
          const float a = Gr[p][rr * 3 + 0];
          const float b = Gr[p][rr * 3 + 1];
          const float c = Gr[p][rr * 3 + 2];
          Gr[i][rr * 3 + 0] = a * r0 + b * r3 + c * r6;
          Gr[i][rr * 3 + 1] = a * r1 + b * r4 + c * r7;
          Gr[i][rr * 3 + 2] = a * r2 + b * r5 + c * r8;
          Gt[i][rr]         = a * rx + b * ry + c * rz + Gt[p][rr];
        }
      }

      // Stage joint position (stride 165 dwords, odd -> conflict-free).
      pOut[tid * 165 + i * 3 + 0] = Gt[i][0];
      pOut[tid * 165 + i * 3 + 1] = Gt[i][1];
      pOut[tid * 165 + i * 3 + 2] = Gt[i][2];
    }

#if USE_TDM
    __syncthreads();  // all waves done reading buf[s&1]; safe to overwrite
    if ((threadIdx.x < 32) && (s + 2 < NSTAGE)) {
      const uint32_t dstLds = ((s & 1) == 0) ? ldsIn0 : ldsIn1;  // (s+2)&1 == s&1
      tdm_load(tdm_g0(dstLds, rotBase + (uint64_t)(s + 2) * 180u),
               tdm_g1(45u, TILE, 45u, TILE, 495u));
    }
#endif
  }

  __syncthreads();  // all sOut ds_stores complete (barrier waits DScnt)
#if USE_TDM
  if (threadIdx.x < 32) {
    // Contiguous 64*165-float tile -> global.
    tdm_store(tdm_g0(ldsOut, outBase), tdm_g1(165u, TILE, 165u, TILE, 165u));
    __builtin_amdgcn_s_wait_tensorcnt(0);
  }
#else
  for (int t = tid; t < N_JOINTS * 3 * TILE; t += TILE)
    out[(size_t)blk * (TILE * 165u) + (size_t)t] = pOut[t];
#endif
}

extern "C" void kernel_launch(void* const* d_in, const int* in_sizes, int n_in,
                              void* d_out, int out_size, void* d_ws, size_t ws_size,
                              hipStream_t stream) {
  (void)n_in; (void)out_size; (void)d_ws; (void)ws_size;
  const float* rot = (const float*)d_in[0];   // [B,55,3,3] fp32
  const float* Jg  = (const float*)d_in[1];   // [55,3] fp32
  float* out       = (float*)d_out;           // [B,55,3] fp32

  const int B      = in_sizes[0] / (N_JOINTS * 9);  // 65536
  const int blocks = B / TILE;                      // 1024

  smplx_rigid_chain_kernel<<<blocks, TILE, 0, stream>>>(rot, Jg, out);
}